// RetinaNetHead_19439021982037
// MI455X (gfx1250) — compile-verified
//
#include <hip/hip_runtime.h>

#define NC 80      // classes
#define NA 9       // anchors per cell
#define NN 8       // batch
#define NH 100
#define NW 100
#define NG 64      // gt boxes per image
#define HW (NH * NW)
#define HWA (HW * NA)
#define W4 (NW / 4)
#define TOTAL (NN * NA * NH * W4)   // 180,000 threads, 4 anchors each

typedef float v2f __attribute__((ext_vector_type(2)));
typedef float v4f __attribute__((ext_vector_type(4)));
typedef float v8f __attribute__((ext_vector_type(8)));

// ---------------- wave-level f32 sum via V_WMMA_F32_16X16X4_F32 -------------
// A (16x4 f32): lane L<16 supplies K=0,1 (vgpr0,1) of row L; lane L>=16 supplies
// K=2,3 of row L-16.  Put the partial in vgpr0, zero in vgpr1; B = ones(4x16).
// D[m][*] = p_m + p_{m+16} (exact f32).  Each lane then sums its 8 D rows and
// one shfl_xor(16) combines rows 0-7 with rows 8-15.
__device__ __forceinline__ float wave_sum_f32(float v) {
#if __has_builtin(__builtin_amdgcn_wmma_f32_16x16x4_f32)
  v2f a; a[0] = v;    a[1] = 0.0f;
  v2f b; b[0] = 1.0f; b[1] = 1.0f;
  v8f c = {};
  v8f d = __builtin_amdgcn_wmma_f32_16x16x4_f32(false, a, false, b,
                                                (short)0, c, false, false);
  float r = ((d[0] + d[1]) + (d[2] + d[3])) + ((d[4] + d[5]) + (d[6] + d[7]));
  r += __shfl_xor(r, 16, 32);
  return r;
#else
  for (int o = 16; o > 0; o >>= 1) v += __shfl_xor(v, o, 32);
  return v;
#endif
}

// ---------------- focal loss term (exact reference math, stable) ------------
__device__ __forceinline__ float focal_term(float x, bool is_t) {
  float em  = __expf(-fabsf(x));            // e^{-|x|}
  float inv = 1.0f / (1.0f + em);           // sigmoid(|x|)
  float sp  = fmaxf(x, 0.0f) + log1pf(em);  // softplus(x) = -log_sigmoid(-x)
  float s   = (x >= 0.0f) ? inv : em * inv; // sigmoid(x)
  float sm  = (x >= 0.0f) ? em * inv : inv; // 1 - sigmoid(x)
  float t0  = 0.75f * s  * s  * sp;         // (1-a) * s^2        * softplus(x)
  float t1  = 0.25f * sm * sm * (sp - x);   //  a    * (1-s)^2    * softplus(-x)
  return is_t ? t1 : t0;
}

__device__ __forceinline__ float box_l1(const v4f* __restrict__ an4,
                                        const v4f* __restrict__ gb4,
                                        int aidx, int gidx,
                                        float q0, float q1, float q2, float q3) {
  v4f an = an4[aidx];
  v4f g  = gb4[gidx];
  float aw = an.z - an.x, ah = an.w - an.y;
  float ax = an.x + 0.5f * aw, ay = an.y + 0.5f * ah;
  float gw = g.z - g.x,   gh = g.w - g.y;
  float gx = g.x + 0.5f * gw, gy = g.y + 0.5f * gh;
  float t0 = (gx - ax) / aw;
  float t1 = (gy - ay) / ah;
  float t2 = logf(gw / aw);
  float t3 = logf(gh / ah);
  return fabsf(q0 - t0) + fabsf(q1 - t1) + fabsf(q2 - t2) + fabsf(q3 - t3);
}

// ---------------- kernel 0: zero accumulators (graph-replay safe) -----------
__global__ void retina_init(int* __restrict__ cnt, float* __restrict__ out) {
  int t = threadIdx.x;
  if (t < NN) cnt[t] = 0;
  if (t == 0) { out[0] = 0.0f; out[1] = 0.0f; }
}

// ---------------- kernel 1: positive count per image ------------------------
__global__ void retina_count(const int* __restrict__ matched,
                             int* __restrict__ cnt) {
  int n     = blockIdx.x >> 3;          // 8 blocks per image
  int chunk = blockIdx.x & 7;
  const int* m = matched + n * HWA;
  int c = 0;
  for (int i = chunk * blockDim.x + threadIdx.x; i < HWA; i += 8 * blockDim.x)
    c += (m[i] >= 0) ? 1 : 0;
  for (int o = 16; o > 0; o >>= 1) c += __shfl_xor(c, o, 32);
  __shared__ int s[8];
  int wave = threadIdx.x >> 5, lane = threadIdx.x & 31;
  if (lane == 0) s[wave] = c;
  __syncthreads();
  if (threadIdx.x == 0) {
    int tot = 0;
    for (int i = 0; i < 8; ++i) tot += s[i];
    atomicAdd(&cnt[n], tot);
  }
}

// ---------------- kernel 2: fused focal + L1 loss, streaming ----------------
__global__ void __launch_bounds__(256)
retina_loss(const float* __restrict__ pc,      // [N, A*C, H, W]
            const float* __restrict__ pb,      // [N, A*4, H, W]
            const float* __restrict__ anchors, // [N, HWA, 4]
            const float* __restrict__ gtb,     // [N, G, 4]
            const int*   __restrict__ glbl,    // [N, G]
            const int*   __restrict__ matched, // [N, HWA]
            const int*   __restrict__ cnt,     // [N]
            float* __restrict__ out) {
  int T = blockIdx.x * 256 + threadIdx.x;
  float partC = 0.0f, partB = 0.0f;

  if (T < TOTAL) {
    int wq = T % W4; int r = T / W4;
    int h  = r % NH; r /= NH;
    int a  = r % NA; int n = r / NA;
    int hw = h * NW + wq * 4;

    const float* pcb = pc + (size_t)((n * NA + a) * NC) * HW + hw;
    const float* pbb = pb + (size_t)((n * NA + a) * 4)  * HW + hw;

    int i0 = hw * NA + a;                    // anchor index of slot 0
    const int* mp = matched + n * HWA + i0;
    int mi0 = mp[0], mi1 = mp[9], mi2 = mp[18], mi3 = mp[27];

    const int* lb = glbl + n * NG;
    bool p0 = mi0 >= 0, p1 = mi1 >= 0, p2 = mi2 >= 0, p3 = mi3 >= 0;
    int tc0 = p0 ? (lb[mi0] - 1) : -1;
    int tc1 = p1 ? (lb[mi1] - 1) : -1;
    int tc2 = p2 ? (lb[mi2] - 1) : -1;
    int tc3 = p3 ? (lb[mi3] - 1) : -1;
    float v0 = (mi0 >= -1) ? 1.0f : 0.0f;
    float v1 = (mi1 >= -1) ? 1.0f : 0.0f;
    float v2 = (mi2 >= -1) ? 1.0f : 0.0f;
    float v3 = (mi3 >= -1) ? 1.0f : 0.0f;

    // ---- focal loss over 80 classes, 4 anchors (w,w+1,w+2,w+3) -----------
    float s0 = 0.f, s1 = 0.f, s2 = 0.f, s3 = 0.f;
#pragma unroll 4
    for (int c = 0; c < NC; ++c) {
      v4f x = __builtin_nontemporal_load((const v4f*)(pcb + c * HW));
      s0 += focal_term(x.x, c == tc0);
      s1 += focal_term(x.y, c == tc1);
      s2 += focal_term(x.z, c == tc2);
      s3 += focal_term(x.w, c == tc3);
    }
    float accC = v0 * s0 + v1 * s1 + v2 * s2 + v3 * s3;

    // ---- L1 box loss on positives ----------------------------------------
    v4f q0 = __builtin_nontemporal_load((const v4f*)(pbb));
    v4f q1 = __builtin_nontemporal_load((const v4f*)(pbb + HW));
    v4f q2 = __builtin_nontemporal_load((const v4f*)(pbb + 2 * HW));
    v4f q3 = __builtin_nontemporal_load((const v4f*)(pbb + 3 * HW));
    const v4f* an4 = (const v4f*)anchors + (size_t)n * HWA;
    const v4f* gb4 = (const v4f*)gtb + (size_t)n * NG;
    float accB = 0.0f;
    if (p0) accB += box_l1(an4, gb4, i0,      mi0, q0.x, q1.x, q2.x, q3.x);
    if (p1) accB += box_l1(an4, gb4, i0 + 9,  mi1, q0.y, q1.y, q2.y, q3.y);
    if (p2) accB += box_l1(an4, gb4, i0 + 18, mi2, q0.z, q1.z, q2.z, q3.z);
    if (p3) accB += box_l1(an4, gb4, i0 + 27, mi3, q0.w, q1.w, q2.w, q3.w);

    int pn = cnt[n];
    float sc = 1.0f / (8.0f * (float)(pn > 0 ? pn : 1)); // 1/(N * pos_num)
    partC = accC * sc;
    partB = accB * sc;
  }

  // ---- block reduction: WMMA wave sum (EXEC all-1s here), then LDS -------
  float rc = wave_sum_f32(partC);
  float rb = wave_sum_f32(partB);
  __shared__ float sC[8], sB[8];
  int wave = threadIdx.x >> 5, lane = threadIdx.x & 31;
  if (lane == 0) { sC[wave] = rc; sB[wave] = rb; }
  __syncthreads();
  if (wave == 0) {
    float c2 = (lane < 8) ? sC[lane] : 0.0f;
    float b2 = (lane < 8) ? sB[lane] : 0.0f;
    for (int o = 4; o > 0; o >>= 1) {
      c2 += __shfl_xor(c2, o, 32);
      b2 += __shfl_xor(b2, o, 32);
    }
    if (lane == 0) {
      atomicAdd(&out[0], c2);
      atomicAdd(&out[1], b2);
    }
  }
}

extern "C" void kernel_launch(void* const* d_in, const int* in_sizes, int n_in,
                              void* d_out, int out_size, void* d_ws, size_t ws_size,
                              hipStream_t stream) {
  const float* pred_class = (const float*)d_in[0];
  const float* pred_bbox  = (const float*)d_in[1];
  const float* anchors    = (const float*)d_in[2];
  const float* gt_boxes   = (const float*)d_in[3];
  const int*   gt_labels  = (const int*)d_in[4];
  const int*   matched    = (const int*)d_in[5];

  int*   cnt = (int*)d_ws;
  float* out = (float*)d_out;

  retina_init<<<1, 32, 0, stream>>>(cnt, out);
  retina_count<<<NN * 8, 256, 0, stream>>>(matched, cnt);
  int blocks = (TOTAL + 255) / 256;
  retina_loss<<<blocks, 256, 0, stream>>>(pred_class, pred_bbox, anchors,
                                          gt_boxes, gt_labels, matched,
                                          cnt, out);
  (void)in_sizes; (void)n_in; (void)out_size; (void)ws_size;
}